// NonMaxSuppression_22127671509740
// MI455X (gfx1250) — compile-verified
//
#include <hip/hip_runtime.h>
#include <stdint.h>

// Problem constants (match reference)
#define Bd 8
#define Hd 2048
#define Wd 2048
#define MAXKP 2000000
#define NROWS_TOTAL (Bd * Hd)       // 16384 row-blocks, ascending == flat order
#define WORDS_PER_ROW (Wd / 32)     // 64 mask words per row

typedef __attribute__((ext_vector_type(4))) unsigned int tdm_v4u;
typedef __attribute__((ext_vector_type(8))) int          tdm_v8i;
typedef __attribute__((ext_vector_type(4))) int          tdm_v4i;

// ---------------------------------------------------------------------------
// K1 (placed first so the disasm snippet shows the TDM sequence):
// per-row 5x5 NMS mask via TDM-staged LDS tile.
//   block = one image row (row_id = b*H + h). TDM DMAs the clamped 5-row
//   window (nrows x 2048 f32, stride 2048) into LDS in one instruction.
// ---------------------------------------------------------------------------
__global__ __launch_bounds__(256) void nms_mask_kernel(
    const float* __restrict__ scores,
    uint32_t* __restrict__ maskWords,
    uint32_t* __restrict__ blockCounts)
{
    __shared__ float tile[5][Wd];         // 40 KB of the 320 KB/WGP LDS
    __shared__ uint32_t s_count;

    const int row_id = blockIdx.x;        // b*H + h
    const int h      = row_id & (Hd - 1);
    const int tid    = threadIdx.x;

    if (tid == 0) s_count = 0;

    const int row_lo  = (h - 2 > 0) ? (h - 2) : 0;
    const int row_hi  = (h + 2 < Hd - 1) ? (h + 2) : (Hd - 1);
    const int nrows   = row_hi - row_lo + 1;
    const int top_pad = row_lo - (h - 2);               // 0..2 missing top rows

    // ---- Tensor Data Mover: global (nrows x 2048 f32) tile -> LDS --------
    if (tid < 32) {                                     // wave 0 issues once
        const uint64_t gaddr =
            (uint64_t)(uintptr_t)(scores + (size_t)(row_id - (h - row_lo)) * Wd);
        const uint32_t lds_addr = (uint32_t)(uintptr_t)(&tile[top_pad][0]);

        // D# group 0 (128b): count=1 | lds_addr | global_addr[56:0] | type=2
        tdm_v4u g0;
        g0.x = 1u;                                      // count=1 (valid), user mode
        g0.y = lds_addr;                                // bits [63:32]
        g0.z = (uint32_t)(gaddr & 0xFFFFFFFFull);       // bits [95:64]
        g0.w = (uint32_t)((gaddr >> 32) & 0x1FFFFFFull) // addr[56:32] -> [120:96]
             | (2u << 30);                              // type=2 ("image") [127:126]

        // D# group 1 (256b)
        const uint32_t tdim0 = Wd;                      // tensor dim0 (elements)
        const uint32_t tdim1 = (uint32_t)nrows;         // tensor dim1
        const uint32_t til0  = Wd;                      // tile dim0
        const uint32_t til1  = (uint32_t)nrows;         // tile dim1
        const uint64_t str0  = Wd;                      // dim0 stride (elements)
        tdm_v8i g1;
        g1[0] = (int)(2u << 16);                        // data_size=2 -> 4 bytes; no barrier/iter/pad
        g1[1] = (int)((tdim0 & 0xFFFFu) << 16);         // tensor_dim0[15:0] @ bits 63:48
        g1[2] = (int)(((tdim0 >> 16) & 0xFFFFu)         // tensor_dim0[31:16] @ 79:64
              |       ((tdim1 & 0xFFFFu) << 16));       // tensor_dim1[15:0] @ 95:80
        g1[3] = (int)(((tdim1 >> 16) & 0xFFFFu)         // tensor_dim1[31:16] @ 111:96
              |       ((til0 & 0xFFFFu) << 16));        // tile_dim0 @ 127:112
        g1[4] = (int)(til1 & 0xFFFFu);                  // tile_dim1 @ 143:128, tile_dim2=0
        g1[5] = (int)(uint32_t)(str0 & 0xFFFFFFFFull);  // dim0_stride[31:0] @ 191:160
        g1[6] = (int)(uint32_t)((str0 >> 32) & 0xFFFFull); // dim0_stride[47:32]; dim1_stride[15:0]=0
        g1[7] = 0;
        tdm_v4i g2 = {0, 0, 0, 0};                      // tile_dim2==0 => dims >=2 unused
        tdm_v4i g3 = {0, 0, 0, 0};
        tdm_v8i g4 = {0, 0, 0, 0, 0, 0, 0, 0};          // extra group (clang-23 form), unused

        __builtin_amdgcn_tensor_load_to_lds(g0, g1, g2, g3, g4, 0);
    }

    // Fill out-of-image halo rows with -inf (safe identity for the window max)
    for (int r = 0; r < 5; ++r) {
        if (r < top_pad || r >= top_pad + nrows) {
            for (int w = tid; w < Wd; w += 256) tile[r][w] = -__builtin_inff();
        }
    }

    if (tid < 32) __builtin_amdgcn_s_wait_tensorcnt(0); // TENSORcnt drain (wave 0)
    __syncthreads();

    // ---- 5x5 window max out of LDS; ballot-pack survivors ----------------
    const int wave = tid >> 5;
    const int lane = tid & 31;
    uint32_t mySum = 0;
#pragma unroll
    for (int k = 0; k < 8; ++k) {
        const int e = wave * 256 + k * 32 + lane;       // column 0..2047
        const float s = tile[2][e];                     // center row is tile[2]
        float m = -__builtin_inff();
#pragma unroll
        for (int r = 0; r < 5; ++r) {
#pragma unroll
            for (int dx = -2; dx <= 2; ++dx) {
                const int x = e + dx;
                const float v = ((unsigned)x < (unsigned)Wd) ? tile[r][x]
                                                             : -__builtin_inff();
                m = fmaxf(m, v);
            }
        }
        const bool pred = (s > 0.0f) && (s == m);       // self in window => s is max
        const uint32_t bits = (uint32_t)__ballot(pred); // wave32 ballot
        if (lane == 0) {
            maskWords[(size_t)row_id * WORDS_PER_ROW + wave * 8 + k] = bits;
            mySum += (uint32_t)__popc(bits);
        }
    }
    if (lane == 0 && mySum) atomicAdd(&s_count, mySum);
    __syncthreads();
    if (tid == 0) blockCounts[row_id] = s_count;
}

// ---------------------------------------------------------------------------
// K0: fill output with -1 (matches jnp.nonzero fill_value=-1 padding)
// ---------------------------------------------------------------------------
__global__ void fill_kernel(int* __restrict__ out, int n) {
    int i = blockIdx.x * blockDim.x + threadIdx.x;
    const int stride = gridDim.x * blockDim.x;
    for (; i < n; i += stride) out[i] = -1;
}

// ---------------------------------------------------------------------------
// K2: exclusive prefix sum over the 16384 per-row counts (single block).
// ---------------------------------------------------------------------------
__global__ __launch_bounds__(256) void scan_kernel(
    const uint32_t* __restrict__ blockCounts,
    uint32_t* __restrict__ blockOffsets)
{
    __shared__ uint32_t part[256];
    const int tid = threadIdx.x;
    const int per = NROWS_TOTAL / 256;                  // 64 counts per thread
    uint32_t sum = 0;
    for (int i = 0; i < per; ++i) sum += blockCounts[tid * per + i];
    part[tid] = sum;
    __syncthreads();
    for (int off = 1; off < 256; off <<= 1) {           // Hillis-Steele inclusive
        uint32_t v = (tid >= off) ? part[tid - off] : 0u;
        __syncthreads();
        part[tid] += v;
        __syncthreads();
    }
    uint32_t base = (tid == 0) ? 0u : part[tid - 1];    // exclusive base
    for (int i = 0; i < per; ++i) {
        blockOffsets[tid * per + i] = base;
        base += blockCounts[tid * per + i];
    }
}

// ---------------------------------------------------------------------------
// K3: ordered scatter of (h, w) into d_out at exact global rank.
//   out[0 .. 2M)   = hh,  out[2M .. 4M) = ww  (pre-filled with -1 by K0)
// ---------------------------------------------------------------------------
__global__ __launch_bounds__(64) void scatter_kernel(
    const uint32_t* __restrict__ maskWords,
    const uint32_t* __restrict__ blockOffsets,
    int* __restrict__ out)
{
    __shared__ uint32_t pref[WORDS_PER_ROW];
    const int row_id = blockIdx.x;
    const int h      = row_id & (Hd - 1);
    const int tid    = threadIdx.x;                     // 0..63 (one word each)

    const uint32_t word = maskWords[(size_t)row_id * WORDS_PER_ROW + tid];
    pref[tid] = (uint32_t)__popc(word);
    __syncthreads();
    if (tid == 0) {                                     // 64-entry serial exclusive scan
        uint32_t run = 0;
        for (int i = 0; i < WORDS_PER_ROW; ++i) {
            const uint32_t c = pref[i];
            pref[i] = run;
            run += c;
        }
    }
    __syncthreads();

    uint32_t idx  = blockOffsets[row_id] + pref[tid];
    uint32_t bits = word;
    while (bits) {
        const int bit = __ffs(bits) - 1;                // ascending w order
        bits &= bits - 1;
        if (idx < MAXKP) {
            out[idx]         = h;
            out[MAXKP + idx] = tid * 32 + bit;
        }
        ++idx;
    }
}

// ---------------------------------------------------------------------------
extern "C" void kernel_launch(void* const* d_in, const int* in_sizes, int n_in,
                              void* d_out, int out_size, void* d_ws, size_t ws_size,
                              hipStream_t stream)
{
    const float* scores = (const float*)d_in[0];
    int* out = (int*)d_out;

    uint8_t* ws = (uint8_t*)d_ws;
    uint32_t* maskWords    = (uint32_t*)ws;                                  // 4 MiB
    uint32_t* blockCounts  = (uint32_t*)(ws + (size_t)NROWS_TOTAL * WORDS_PER_ROW * 4);
    uint32_t* blockOffsets = blockCounts + NROWS_TOTAL;                      // +64 KiB each

    fill_kernel<<<2048, 256, 0, stream>>>(out, out_size);
    nms_mask_kernel<<<NROWS_TOTAL, 256, 0, stream>>>(scores, maskWords, blockCounts);
    scan_kernel<<<1, 256, 0, stream>>>(blockCounts, blockOffsets);
    scatter_kernel<<<NROWS_TOTAL, 64, 0, stream>>>(maskWords, blockOffsets, out);
}